// MultiOutputLSTM_76459007803398
// MI455X (gfx1250) — compile-verified
//
#include <hip/hip_runtime.h>
#include <cstddef>

typedef float v2f __attribute__((ext_vector_type(2)));
typedef float v8f __attribute__((ext_vector_type(8)));

#define TT      100   // timesteps
#define DD      16    // input features
#define HIDN    64
#define OUTN    2
#define ROWS    64    // batch rows per block
#define NTHREADS 64   // 2 waves (wave32)
#define HSTRIDE 101   // odd stride -> conflict-free column reads
#define SSTRIDE 68    // staging stride

__device__ __forceinline__ float sigmoid_f(float x) {
    return 1.0f / (1.0f + __expf(-x));
}
__device__ __forceinline__ float tanh_f(float x) {
    float ax = fabsf(x);
    float e  = __expf(-2.0f * ax);        // e <= 1, no overflow
    float t  = (1.0f - e) / (1.0f + e);
    return copysignf(t, x);
}

__global__ __launch_bounds__(NTHREADS)
void MultiOutputLSTM_kernel(const float* __restrict__ X,
                            const float* __restrict__ W_ih,
                            const float* __restrict__ W_hh,
                            const float* __restrict__ b_ih,
                            const float* __restrict__ b_hh,
                            const float* __restrict__ W1,
                            const float* __restrict__ b1,
                            const float* __restrict__ W2,
                            const float* __restrict__ b2,
                            const float* __restrict__ W3,
                            const float* __restrict__ b3,
                            float* __restrict__ out)
{
    __shared__ float hbuf[ROWS * HSTRIDE];        // relu(h_t), [row][t]
    __shared__ float stage[2 * 16 * SSTRIDE];     // per-wave 16x64 re-layout tile

    const int tid = threadIdx.x;
    const int g0  = blockIdx.x * ROWS;

    // ---------------- Phase 1: scalar LSTM (hidden size 1), one thread per row
    {
        float wih[4][16];
        #pragma unroll
        for (int gt = 0; gt < 4; ++gt)
            #pragma unroll
            for (int d = 0; d < 16; ++d)
                wih[gt][d] = W_ih[gt * 16 + d];

        float whh[4], bs[4];
        #pragma unroll
        for (int gt = 0; gt < 4; ++gt) {
            whh[gt] = W_hh[gt];
            bs[gt]  = b_ih[gt] + b_hh[gt];
        }

        const int row = g0 + tid;
        const float4* xp = (const float4*)(X + (size_t)row * TT * DD);
        float h = 0.0f, c = 0.0f;

        for (int t = 0; t < TT; ++t) {
            __builtin_prefetch((const char*)(xp + 8), 0, 0);  // ~2 steps ahead -> global_prefetch_b8
            float4 x0 = xp[0], x1 = xp[1], x2 = xp[2], x3 = xp[3];
            xp += 4;
            float xs[16] = { x0.x, x0.y, x0.z, x0.w,
                             x1.x, x1.y, x1.z, x1.w,
                             x2.x, x2.y, x2.z, x2.w,
                             x3.x, x3.y, x3.z, x3.w };
            float gate[4];
            #pragma unroll
            for (int gt = 0; gt < 4; ++gt) {
                float a = fmaf(h, whh[gt], bs[gt]);
                #pragma unroll
                for (int d = 0; d < 16; ++d)
                    a = fmaf(wih[gt][d], xs[d], a);
                gate[gt] = a;
            }
            float ig = sigmoid_f(gate[0]);
            float fg = sigmoid_f(gate[1]);
            float gg = tanh_f(gate[2]);
            float og = sigmoid_f(gate[3]);
            c = fmaf(fg, c, ig * gg);
            h = og * tanh_f(c);
            hbuf[tid * HSTRIDE + t] = fmaxf(h, 0.0f);   // relu(h) for the MLP
        }
    }

    __syncthreads();

    // ---------------- Phase 2: MLP via fp32 WMMA (16x16x4), per-wave 16-row tiles
    const int wv   = tid >> 5;
    const int lane = tid & 31;
    const int half = lane >> 4;   // 0: K pair (k0,k0+1) / M rows 0-7 ; 1: (k0+2,k0+3) / rows 8-15
    const int l16  = lane & 15;
    float* st = stage + wv * (16 * SSTRIDE);

    for (int tt = 0; tt < 2; ++tt) {
        const int rb = (wv * 2 + tt) * 16;   // tile row base within block

        // ---- Layer 1: (16x100) @ (100x64), K in chunks of 4
        v8f acc[4];
        #pragma unroll
        for (int j = 0; j < 4; ++j)
            #pragma unroll
            for (int r = 0; r < 8; ++r) acc[j][r] = 0.0f;

        for (int k0 = 0; k0 < TT; k0 += 4) {
            const int ka = k0 + half * 2;
            v2f a;
            a.x = hbuf[(rb + l16) * HSTRIDE + ka];
            a.y = hbuf[(rb + l16) * HSTRIDE + ka + 1];
            #pragma unroll
            for (int j = 0; j < 4; ++j) {
                v2f bm;   // B[k][n] = W1[n][k]
                bm.x = W1[(j * 16 + l16) * TT + ka];
                bm.y = W1[(j * 16 + l16) * TT + ka + 1];
                acc[j] = __builtin_amdgcn_wmma_f32_16x16x4_f32(
                    false, a, false, bm, (short)0, acc[j], false, false);
            }
        }
        // bias + relu, re-layout C->A via per-wave LDS staging
        #pragma unroll
        for (int j = 0; j < 4; ++j) {
            float bb = b1[j * 16 + l16];
            #pragma unroll
            for (int r = 0; r < 8; ++r)
                st[(half * 8 + r) * SSTRIDE + j * 16 + l16] =
                    fmaxf(acc[j][r] + bb, 0.0f);
        }

        // ---- Layer 2: (16x64) @ (64x64)
        v8f acc2[4];
        #pragma unroll
        for (int j = 0; j < 4; ++j)
            #pragma unroll
            for (int r = 0; r < 8; ++r) acc2[j][r] = 0.0f;

        for (int k0 = 0; k0 < HIDN; k0 += 4) {
            const int ka = k0 + half * 2;
            v2f a;
            a.x = st[l16 * SSTRIDE + ka];
            a.y = st[l16 * SSTRIDE + ka + 1];
            #pragma unroll
            for (int j = 0; j < 4; ++j) {
                v2f bm;   // B[k][n] = W2[n][k]
                bm.x = W2[(j * 16 + l16) * HIDN + ka];
                bm.y = W2[(j * 16 + l16) * HIDN + ka + 1];
                acc2[j] = __builtin_amdgcn_wmma_f32_16x16x4_f32(
                    false, a, false, bm, (short)0, acc2[j], false, false);
            }
        }
        #pragma unroll
        for (int j = 0; j < 4; ++j) {
            float bb = b2[j * 16 + l16];
            #pragma unroll
            for (int r = 0; r < 8; ++r)
                st[(half * 8 + r) * SSTRIDE + j * 16 + l16] =
                    fmaxf(acc2[j][r] + bb, 0.0f);
        }

        // ---- Layer 3 (N=2) + log_softmax: lane = row(l16) x class(half)
        {
            const float* w3 = W3 + half * HIDN;
            float z = b3[half];
            #pragma unroll
            for (int k = 0; k < HIDN; ++k)
                z = fmaf(st[l16 * SSTRIDE + k], w3[k], z);

            float zo = __shfl_xor(z, 16, 32);      // the other class, wave32
            float mx = fmaxf(z, zo);
            float s  = __expf(z - mx) + __expf(zo - mx);
            float res = (z - mx) - __logf(s);

            const size_t row = (size_t)(g0 + rb + l16);
            out[row * OUTN + half] = res;
        }
    }
}

extern "C" void kernel_launch(void* const* d_in, const int* in_sizes, int n_in,
                              void* d_out, int out_size, void* d_ws, size_t ws_size,
                              hipStream_t stream) {
    (void)n_in; (void)d_ws; (void)ws_size; (void)out_size;
    const float* X    = (const float*)d_in[0];
    const float* W_ih = (const float*)d_in[1];
    const float* W_hh = (const float*)d_in[2];
    const float* b_ih = (const float*)d_in[3];
    const float* b_hh = (const float*)d_in[4];
    const float* W1   = (const float*)d_in[5];
    const float* b1   = (const float*)d_in[6];
    const float* W2   = (const float*)d_in[7];
    const float* b2   = (const float*)d_in[8];
    const float* W3   = (const float*)d_in[9];
    const float* b3   = (const float*)d_in[10];
    float* out = (float*)d_out;

    const int Bn = in_sizes[0] / (TT * DD);     // 65536
    const int grid = Bn / ROWS;                 // 1024 blocks, 64 rows each

    MultiOutputLSTM_kernel<<<grid, NTHREADS, 0, stream>>>(
        X, W_ih, W_hh, b_ih, b_hh, W1, b1, W2, b2, W3, b3, out);
}